// Interaction_Net_Model_49555332662129
// MI455X (gfx1250) — compile-verified
//
#include <hip/hip_runtime.h>

typedef __attribute__((ext_vector_type(2))) float v2f;
typedef __attribute__((ext_vector_type(8))) float v8f;

#define DD 32

// D = A(16x4 f32) * B(4x16 f32) + C(16x16 f32), wave32 WMMA
static __device__ __forceinline__ v8f wmma4(v2f a, v2f b, v8f c) {
    return __builtin_amdgcn_wmma_f32_16x16x4_f32(
        /*neg_a=*/false, a, /*neg_b=*/false, b,
        /*c_mod=*/(short)0, c, /*reuse_a=*/false, /*reuse_b=*/false);
}

// A fragment (16x4, f32) at k0 from a row-major [16, 32] tile starting at Xrow0.
// Lane layout: lanes 0-15 hold M=lane, K=k0..k0+1; lanes 16-31 hold M=lane-16, K=k0+2..k0+3.
static __device__ __forceinline__ v2f afrag(const float* Xrow0, int k0, int lane) {
    int m = lane & 15;
    int k = k0 + ((lane >> 4) << 1);
    const float* p = Xrow0 + m * DD + k;
    v2f a; a.x = p[0]; a.y = p[1];
    return a;
}

// B fragment (4x16, f32) at (k0, n0) for computing X @ W^T with W row-major [32,32]:
// B[k][n] = W[n][k]. Mirrors the A-matrix half-wave K split.
static __device__ __forceinline__ v2f bfrag(const float* W, int k0, int n0, int lane) {
    int n = n0 + (lane & 15);
    int k = k0 + ((lane >> 4) << 1);
    const float* p = W + n * DD + k;
    v2f b; b.x = p[0]; b.y = p[1];
    return b;
}

// Store 16x16 f32 D tile (8 VGPRs): VGPR r -> row (r + (lane>>4)*8), col n0 + (lane&15)
static __device__ __forceinline__ void store_d(float* Orow0, int n0, int lane, v8f c) {
    int n = n0 + (lane & 15);
    int mb = (lane >> 4) << 3;
#pragma unroll
    for (int r = 0; r < 8; ++r) Orow0[(size_t)(mb + r) * DD + n] = c[r];
}

// ---------------- K1: node embeddings (3 GEMMs) + zero the scatter accumulator ----------------
__global__ __launch_bounds__(256) void k_nodes(
    const float* __restrict__ x,
    const float* __restrict__ Wsrc, const float* __restrict__ Wrx,
    const float* __restrict__ Wrxnode,
    float* __restrict__ ws_src, float* __restrict__ ws_rx,
    float* __restrict__ out_rxnode, float* __restrict__ ws_aggr, int ntiles) {
    int lane = threadIdx.x & 31;
    int tile = blockIdx.x * (blockDim.x >> 5) + (threadIdx.x >> 5);
    if (tile >= ntiles) return;
    size_t row0 = (size_t)tile * 16;
    const float* xb = x + row0 * DD;

    v2f a[8];
#pragma unroll
    for (int j = 0; j < 8; ++j) a[j] = afrag(xb, 4 * j, lane);

    const float* Ws[3] = {Wsrc, Wrx, Wrxnode};
    float* Os[3] = {ws_src + row0 * DD, ws_rx + row0 * DD, out_rxnode + row0 * DD};
#pragma unroll
    for (int w = 0; w < 3; ++w) {
#pragma unroll
        for (int nt = 0; nt < 2; ++nt) {
            v8f c = {};
#pragma unroll
            for (int j = 0; j < 8; ++j) c = wmma4(a[j], bfrag(Ws[w], 4 * j, nt * 16, lane), c);
            store_d(Os[w], nt * 16, lane, c);
        }
    }

    // zero this tile's slice of the scatter accumulator (512 floats) with b128 stores
    float4 z; z.x = z.y = z.z = z.w = 0.0f;
    float4* zp = (float4*)(ws_aggr + row0 * DD);
#pragma unroll
    for (int i = 0; i < 4; ++i) zp[i * 32 + lane] = z;
}

// ---------------- K2: fused edge pipeline ----------------
// agg = src_embed[u] + rx_embed[v]  (gather -> C init, served from 192MB L2)
// agg += edge_attr @ Wedge^T        (WMMA, streaming NT loads of edge_attr)
// upd  = agg @ Wupd^T               (D-layout -> A-layout transpose via per-wave LDS)
// ws_aggr[v] += upd                 (relaxed agent-scope f32 atomics)
__global__ __launch_bounds__(256) void k_edges(
    const int* __restrict__ eidx,          // [2*E], row0 = u, row1 = v
    const float* __restrict__ edge_attr,   // [E, 32]
    const float* __restrict__ Wedge, const float* __restrict__ Wupd,
    const float* __restrict__ ws_src, const float* __restrict__ ws_rx,
    float* __restrict__ ws_aggr, int Ecount, int ntiles) {
    __shared__ float lds[8][16 * 33];      // per-wave 16x32 tile, padded stride 33
    int lane = threadIdx.x & 31;
    int wid = threadIdx.x >> 5;
    int tile = blockIdx.x * 8 + wid;
    if (tile >= ntiles) return;            // no block-wide barriers below: safe
    size_t e0 = (size_t)tile * 16;

    const int* uP = eidx;
    const int* vP = eidx + Ecount;
    int ncol = lane & 15;
    int mb = (lane >> 4) << 3;             // rows this lane covers in C/D layout

    int uu[8], vv[8];
#pragma unroll
    for (int r = 0; r < 8; ++r) {
        uu[r] = __builtin_nontemporal_load(uP + e0 + mb + r);
        vv[r] = __builtin_nontemporal_load(vP + e0 + mb + r);
    }

    // C init: gathered src_embed[u] + rx_embed[v] (hits L2-resident tables)
    v8f c0 = {}, c1 = {};
#pragma unroll
    for (int r = 0; r < 8; ++r) {
        const float* su = ws_src + (size_t)uu[r] * DD;
        const float* rv = ws_rx + (size_t)vv[r] * DD;
        c0[r] = su[ncol] + rv[ncol];
        c1[r] = su[ncol + 16] + rv[ncol + 16];
    }

    // A fragments from edge_attr, non-temporal (256MB read-once stream; don't pollute L2)
    const float* eb = edge_attr + e0 * DD;
    v2f a[8];
    {
        int m = lane & 15;
        int koff = (lane >> 4) << 1;
#pragma unroll
        for (int j = 0; j < 8; ++j) {
            const float* p = eb + m * DD + 4 * j + koff;
            a[j].x = __builtin_nontemporal_load(p);
            a[j].y = __builtin_nontemporal_load(p + 1);
        }
    }

#pragma unroll
    for (int j = 0; j < 8; ++j) c0 = wmma4(a[j], bfrag(Wedge, 4 * j, 0, lane), c0);
#pragma unroll
    for (int j = 0; j < 8; ++j) c1 = wmma4(a[j], bfrag(Wedge, 4 * j, 16, lane), c1);

    // D-layout -> A-layout through per-wave LDS (same-wave DS ops are in-order)
    float* L = &lds[wid][0];
#pragma unroll
    for (int r = 0; r < 8; ++r) {
        L[(mb + r) * 33 + ncol] = c0[r];
        L[(mb + r) * 33 + ncol + 16] = c1[r];
    }
    v2f a2[8];
    {
        int m = lane & 15;
        int koff = (lane >> 4) << 1;
#pragma unroll
        for (int j = 0; j < 8; ++j) {
            int k = 4 * j + koff;
            a2[j].x = L[m * 33 + k];
            a2[j].y = L[m * 33 + k + 1];
        }
    }

    v8f d0 = {}, d1 = {};
#pragma unroll
    for (int j = 0; j < 8; ++j) d0 = wmma4(a2[j], bfrag(Wupd, 4 * j, 0, lane), d0);
#pragma unroll
    for (int j = 0; j < 8; ++j) d1 = wmma4(a2[j], bfrag(Wupd, 4 * j, 16, lane), d1);

    // segment-sum scatter: relaxed agent-scope f32 atomic add (L2 atomic units)
#pragma unroll
    for (int r = 0; r < 8; ++r) {
        float* p = ws_aggr + (size_t)vv[r] * DD;
        __hip_atomic_fetch_add(p + ncol, d0[r], __ATOMIC_RELAXED, __HIP_MEMORY_SCOPE_AGENT);
        __hip_atomic_fetch_add(p + ncol + 16, d1[r], __ATOMIC_RELAXED, __HIP_MEMORY_SCOPE_AGENT);
    }
}

// ---------------- K3: rx_aggregated_edges = aggregated @ Waggr^T ----------------
__global__ __launch_bounds__(256) void k_aggr(
    const float* __restrict__ ws_aggr, const float* __restrict__ Waggr,
    float* __restrict__ out2, int ntiles) {
    int lane = threadIdx.x & 31;
    int tile = blockIdx.x * (blockDim.x >> 5) + (threadIdx.x >> 5);
    if (tile >= ntiles) return;
    size_t row0 = (size_t)tile * 16;
    const float* ab = ws_aggr + row0 * DD;

    v2f a[8];
#pragma unroll
    for (int j = 0; j < 8; ++j) a[j] = afrag(ab, 4 * j, lane);
#pragma unroll
    for (int nt = 0; nt < 2; ++nt) {
        v8f c = {};
#pragma unroll
        for (int j = 0; j < 8; ++j) c = wmma4(a[j], bfrag(Waggr, 4 * j, nt * 16, lane), c);
        store_d(out2 + row0 * DD, nt * 16, lane, c);
    }
}

extern "C" void kernel_launch(void* const* d_in, const int* in_sizes, int n_in,
                              void* d_out, int out_size, void* d_ws, size_t ws_size,
                              hipStream_t stream) {
    const float* x         = (const float*)d_in[0];
    const int*   eidx      = (const int*)d_in[1];      // edge_index [2,E]
    const float* edge_attr = (const float*)d_in[2];
    const float* Wsrc      = (const float*)d_in[3];
    const float* Wedge     = (const float*)d_in[4];
    const float* Wrx       = (const float*)d_in[5];
    const float* Wupd      = (const float*)d_in[6];
    const float* Wrxnode   = (const float*)d_in[7];
    const float* Waggr     = (const float*)d_in[8];

    int Nn = in_sizes[0] / DD;   // 100000
    int Ee = in_sizes[1] / 2;    // 2000000

    float* out_rxnode = (float*)d_out;
    float* out_aggr   = (float*)d_out + (size_t)Nn * DD;

    float* ws_src  = (float*)d_ws;                      // [N,32]
    float* ws_rx   = ws_src + (size_t)Nn * DD;          // [N,32]
    float* ws_aggr = ws_rx + (size_t)Nn * DD;           // [N,32]  (~38.4 MB total)

    int ntilesN = (Nn + 15) / 16;                       // 6250 (exact)
    int ntilesE = (Ee + 15) / 16;                       // 125000 (exact)
    int nblkN = (ntilesN + 7) / 8;
    int nblkE = (ntilesE + 7) / 8;

    k_nodes<<<nblkN, 256, 0, stream>>>(x, Wsrc, Wrx, Wrxnode,
                                       ws_src, ws_rx, out_rxnode, ws_aggr, ntilesN);
    k_edges<<<nblkE, 256, 0, stream>>>(eidx, edge_attr, Wedge, Wupd,
                                       ws_src, ws_rx, ws_aggr, Ee, ntilesE);
    k_aggr<<<nblkN, 256, 0, stream>>>(ws_aggr, Waggr, out_aggr, ntilesN);
}